// HausdorffLoss_46377056862526
// MI455X (gfx1250) — compile-verified
//
#include <hip/hip_runtime.h>
#include <math.h>

typedef __attribute__((ext_vector_type(2))) float v2f;
typedef __attribute__((ext_vector_type(8))) float v8f;

#define WAVE 32

__global__ void haus_init_ws(unsigned int* ws) {
    ws[0] = 0u;
    ws[1] = 0u;
}

// One wave handles 16 "query" points of one batch and sweeps all "target"
// points of that batch in 16-wide tiles via V_WMMA_F32_16X16X4_F32.
// A row i = (x,y,z,1); B col j = (-2x,-2y,-2z,||t_j||^2)
//   => D[i][j] = ||t_j||^2 - 2<q_i,t_j>;  dist^2 = D[i][j] + ||q_i||^2
// Reduces min over targets per query, then max over queries, atomicMax(ws).
__global__ __launch_bounds__(256)
void hausdorff_dir(const float* __restrict__ q, const float* __restrict__ t,
                   int nQ, int nT, int tilesPerBatch, unsigned int* ws) {
    const int lane = threadIdx.x & 31;
    const int wave = threadIdx.x >> 5;
    const int strip = blockIdx.x * 8 + wave;       // global 16-row strip id
    const int b = strip / tilesPerBatch;
    const int rowTile = strip - b * tilesPerBatch;
    const int hi  = lane >> 4;                     // which 16-lane half
    const int col = lane & 15;

    const float* qb = q + (size_t)b * nQ * 3;
    const float* tb = t + (size_t)b * nT * 3;

    // ---- A tile (persistent across the sweep) ----
    // f32 16x4 A layout: lane l: row = l%16; VGPR v holds K = v + 2*(l>=16)
    {
    }
    const int r = rowTile * 16 + col;
    const float qx = qb[3 * r + 0];
    const float qy = qb[3 * r + 1];
    const float qz = qb[3 * r + 2];
    v2f a;
    a.x = hi ? qz   : qx;   // K=2 : K=0
    a.y = hi ? 1.0f : qy;   // K=3 : K=1

    float acc[8];
#pragma unroll
    for (int i = 0; i < 8; ++i) acc[i] = __builtin_inff();

    const int nTileT = nT >> 4;
    for (int tt = 0; tt < nTileT; ++tt) {
        // ---- B tile ----
        // f32 4x16 B layout: lane l: col = l%16; VGPR v holds K = v + 2*(l>=16)
        const int j = tt * 16 + col;
        const float tx = tb[3 * j + 0];
        const float ty = tb[3 * j + 1];
        const float tz = tb[3 * j + 2];
        const float tsq = tx * tx + ty * ty + tz * tz;
        v2f bm;
        bm.x = hi ? (-2.0f * tz) : (-2.0f * tx);   // K=2 : K=0
        bm.y = hi ? tsq          : (-2.0f * ty);   // K=3 : K=1

        v8f c = {};
        v8f d = __builtin_amdgcn_wmma_f32_16x16x4_f32(
            /*neg_a=*/false, a, /*neg_b=*/false, bm,
            /*c_mod=*/(short)0, c, /*reuse_a=*/false, /*reuse_b=*/false);

#pragma unroll
        for (int i = 0; i < 8; ++i) acc[i] = fminf(acc[i], d[i]);
    }

    // Min across the 16 columns: xor masks 1..8 stay inside each half-wave.
    // After this, half0 lanes hold min of row v, half1 lanes min of row v+8.
#pragma unroll
    for (int i = 0; i < 8; ++i) {
        float m = acc[i];
        m = fminf(m, __shfl_xor(m, 1, WAVE));
        m = fminf(m, __shfl_xor(m, 2, WAVE));
        m = fminf(m, __shfl_xor(m, 4, WAVE));
        m = fminf(m, __shfl_xor(m, 8, WAVE));
        acc[i] = m;
    }

    // ||q||^2 of point (rowTile*16 + (lane&7) + 8*hi), gathered so that each
    // lane can add the row-matching value for all of its 8 accumulator rows.
    const int myr = rowTile * 16 + (lane & 7) + 8 * hi;
    const float px = qb[3 * myr + 0];
    const float py = qb[3 * myr + 1];
    const float pz = qb[3 * myr + 2];
    const float myq = px * px + py * py + pz * pz;

    float lanemax = 0.0f;
#pragma unroll
    for (int v = 0; v < 8; ++v) {
        const float qsq = __shfl(myq, (hi << 4) + v, WAVE); // ||q_{v+8*hi}||^2
        const float cand = acc[v] + qsq;   // min dist^2 for row v + 8*hi
        lanemax = fmaxf(lanemax, cand);
    }
    // Max over the full wave (covers all 16 rows; duplicates are harmless).
    lanemax = fmaxf(lanemax, __shfl_xor(lanemax, 1, WAVE));
    lanemax = fmaxf(lanemax, __shfl_xor(lanemax, 2, WAVE));
    lanemax = fmaxf(lanemax, __shfl_xor(lanemax, 4, WAVE));
    lanemax = fmaxf(lanemax, __shfl_xor(lanemax, 8, WAVE));
    lanemax = fmaxf(lanemax, __shfl_xor(lanemax, 16, WAVE));
    lanemax = fmaxf(lanemax, 0.0f);  // keep uint-ordered atomic valid

    if (lane == 0)
        atomicMax(ws, __float_as_uint(lanemax));
}

__global__ void haus_finish(const unsigned int* __restrict__ ws,
                            float* __restrict__ out) {
    out[0] = sqrtf(__uint_as_float(ws[0])) + sqrtf(__uint_as_float(ws[1]));
}

extern "C" void kernel_launch(void* const* d_in, const int* in_sizes, int n_in,
                              void* d_out, int out_size, void* d_ws, size_t ws_size,
                              hipStream_t stream) {
    const float* pred  = (const float*)d_in[0];   // [B, N, 3] f32
    const float* label = (const float*)d_in[1];   // [B, M, 3] f32
    float* out = (float*)d_out;                   // scalar f32
    unsigned int* ws = (unsigned int*)d_ws;

    const int B = 4;
    const int N = in_sizes[0] / (B * 3);          // 4096
    const int M = in_sizes[1] / (B * 3);          // 4096

    haus_init_ws<<<1, 1, 0, stream>>>(ws);

    // dis_xy: for each pred point, min over labels; max over all pred points.
    {
        const int strips = B * (N / 16);
        hausdorff_dir<<<dim3(strips / 8), 256, 0, stream>>>(
            pred, label, N, M, N / 16, ws + 0);
    }
    // dis_yx: for each label point, min over preds; max over all label points.
    {
        const int strips = B * (M / 16);
        hausdorff_dir<<<dim3(strips / 8), 256, 0, stream>>>(
            label, pred, M, N, M / 16, ws + 1);
    }

    haus_finish<<<1, 1, 0, stream>>>(ws, out);
}